// ATTN_88622355185799
// MI455X (gfx1250) — compile-verified
//
#include <hip/hip_runtime.h>
#include <math.h>
#include <stdint.h>

#define BS 8
#define NN 160
#define DX 256
#define NH 8
#define DE 64
#define DF 32

typedef __attribute__((ext_vector_type(2))) float v2f;
typedef __attribute__((ext_vector_type(8))) float v8f;
typedef uint32_t u32;
typedef __attribute__((ext_vector_type(4))) u32 u32x4;
typedef __attribute__((ext_vector_type(8))) u32 u32x8;

#define EPAD 68  // E row stride in LDS: 64 data + 4 pad dwords (TDM pad_amount=3)

__device__ __forceinline__ v8f wmma_f32_k4(v2f a, v2f b, v8f c) {
  // D = A(16x4,f32) * B(4x16,f32) + C(16x16,f32)
  return __builtin_amdgcn_wmma_f32_16x16x4_f32(false, a, false, b, (short)0, c,
                                               false, false);
}

__device__ __forceinline__ float blockSum256(float v, float* red) {
#pragma unroll
  for (int off = 16; off; off >>= 1) v += __shfl_xor(v, off, 32);
  __syncthreads();  // protect red[] from any previous use
  if ((threadIdx.x & 31) == 0) red[threadIdx.x >> 5] = v;
  __syncthreads();
  float tot = 0.f;
#pragma unroll
  for (int w = 0; w < 8; ++w) tot += red[w];
  return tot;
}

// ---------------------------------------------------------------------------
// Kernel 1: Q/K/V projections, f32 WMMA 16x16x4, fused bias + atom mask.
// grid = 80 blocks (b * 10 row-tiles), 256 threads (8 waves).
// ---------------------------------------------------------------------------
__global__ __launch_bounds__(256) void qkv_kernel(
    const float* __restrict__ X, const float* __restrict__ amask,
    const float* __restrict__ Wq, const float* __restrict__ bq,
    const float* __restrict__ Wk, const float* __restrict__ bk,
    const float* __restrict__ Wv, const float* __restrict__ bv,
    float* __restrict__ Qo, float* __restrict__ Ko, float* __restrict__ Vo) {
  const int b = blockIdx.x / (NN / 16);
  const int it = blockIdx.x % (NN / 16);
  __shared__ float Xs[16][DX + 1];  // +1 pad: conflict-free column reads
  const int tid = threadIdx.x;

  const float* xsrc = X + ((size_t)b * NN + it * 16) * DX;
  for (int idx = tid; idx < 16 * DX; idx += 256)
    Xs[idx >> 8][idx & (DX - 1)] = xsrc[idx];
  __syncthreads();

  const int wave = tid >> 5, lane = tid & 31;
  const int lhalf = lane >> 4, l16 = lane & 15;

  // 48 output tiles: 3 matrices x 16 column tiles of 16
  for (int t = wave; t < 48; t += 8) {
    const int m = t >> 4, ct = t & 15;
    const float* W = (m == 0) ? Wq : (m == 1) ? Wk : Wv;
    const float* bia = (m == 0) ? bq : (m == 1) ? bk : bv;
    float* O = (m == 0) ? Qo : (m == 1) ? Ko : Vo;
    const int ncol = ct * 16 + l16;
    const float* wrow = W + (size_t)ncol * DX;  // B[k,c] = W[c,k]
    v8f acc = {};
#pragma unroll 8
    for (int kk = 0; kk < DX / 4; ++kk) {
      const int k0 = kk * 4 + lhalf * 2;
      v2f a, bf;
      a.x = Xs[l16][k0];
      a.y = Xs[l16][k0 + 1];
      bf.x = wrow[k0];
      bf.y = wrow[k0 + 1];
      acc = wmma_f32_k4(a, bf, acc);
    }
    const float bias = bia[ncol];
#pragma unroll
    for (int r = 0; r < 8; ++r) {
      const int row = it * 16 + r + lhalf * 8;  // C/D layout: M = r (+8 hi half)
      O[((size_t)b * NN + row) * DX + ncol] =
          (acc[r] + bias) * amask[b * NN + row];
    }
  }
}

// ---------------------------------------------------------------------------
// Kernel 2: per-(b,i) fused edge-proj (WMMA, E staged via TDM async DMA) +
//           scores + softmax-diag + LN1 + Wo + ReLU + LN2.
// grid = (160, 8), 256 threads (8 waves).
// ---------------------------------------------------------------------------
__global__ __launch_bounds__(256) void attn_kernel(
    const float* __restrict__ X, const float* __restrict__ E,
    const float* __restrict__ bmask, const float* __restrict__ Q,
    const float* __restrict__ K, const float* __restrict__ V,
    const float* __restrict__ Wp0, const float* __restrict__ bp0,
    const float* __restrict__ Wp1, const float* __restrict__ bp1,
    const float* __restrict__ g1, const float* __restrict__ b1,
    const float* __restrict__ Wo, const float* __restrict__ bo,
    const float* __restrict__ g2, const float* __restrict__ b2,
    float* __restrict__ out) {
  const int i = blockIdx.x;
  const int b = blockIdx.y;
  const int tid = threadIdx.x;
  const int wave = tid >> 5, lane = tid & 31;
  const int lhalf = lane >> 4, l16 = lane & 15;

  __shared__ float Es[NN][EPAD];      // E tile, TDM-padded rows (bank-free)
  __shared__ float Wp0s[DF][DE + 1];  // stride 65 -> conflict-free
  __shared__ float p0s[NN][DF + 1];   // stride 33
  __shared__ float scs[NN * 9];       // stride 9
  __shared__ float Qs[NH][DF + 1];
  __shared__ float y1[DX];
  __shared__ float red[8];
  __shared__ float adiag[NH];
  __shared__ float p1d[DF];

  const float* Eb = E + (((size_t)b * NN + i) * NN) * DE;
  const float* bmrow = bmask + ((size_t)b * NN + i) * NN;

  // --- Tensor Data Mover: async DMA E[b,i,:,:] (160x64 f32) -> LDS, with
  //     hardware padding of 4 dwords every 64 dwords (row stride = 68 floats).
  //     Issued once by wave 0; D# built per CDNA5 ISA Ch.8 (2D tensor,
  //     groups 2/3 = NULL).
  if (tid < 32) {
    const uint64_t ga = (uint64_t)(uintptr_t)Eb;
    // Generic LDS pointers carry the byte offset in the low 32 bits.
    const u32 ldsa = (u32)(uintptr_t)(void*)&Es[0][0];
    u32x4 g0;
    g0[0] = 1u;                                       // count=1 (valid D#)
    g0[1] = ldsa;                                     // lds_addr
    g0[2] = (u32)ga;                                  // global_addr[31:0]
    g0[3] = (u32)((ga >> 32) & 0x1FFFFFFu) | (2u << 30);  // addr[56:32]|type=2
    u32x8 g1v;
    g1v[0] = (2u << 16)      // data_size = 4B
           | (1u << 20)      // pad_enable
           | (5u << 22)      // pad_interval code 5 = every 64 dwords
           | (3u << 25);     // pad_amount  code 3 = 4 dwords
    g1v[1] = (64u << 16);    // tensor_dim0 = 64
    g1v[2] = (160u << 16);   // tensor_dim1 = 160
    g1v[3] = (64u << 16);    // tile_dim0 = 64
    g1v[4] = 160u;           // tile_dim1 = 160 (tile_dim2 = 0)
    g1v[5] = 64u;            // tensor_dim0_stride = 64
    g1v[6] = (10240u << 16); // tensor_dim1_stride = 160*64 (low 16 bits)
    g1v[7] = 0u;
    asm volatile("tensor_load_to_lds %0, %1" ::"s"(g0), "s"(g1v) : "memory");
  }

  // Overlap with the DMA: stage Wp0 and the Q row.
  for (int idx = tid; idx < DF * DE; idx += 256)
    Wp0s[idx >> 6][idx & 63] = Wp0[idx];
  Qs[tid >> 5][tid & 31] = Q[((size_t)b * NN + i) * DX + tid];

  if (tid < 32) __builtin_amdgcn_s_wait_tensorcnt(0);
  __syncthreads();

  // p0[j,d] = tanh(E[b,i,j,:] @ Wp0[d,:] + bp0[d]) * bond_mask[b,i,j]
  // Both 16-wide d-halves share the same A fragment -> E touched once.
  for (int jt = wave; jt < NN / 16; jt += 8) {
    v8f acc0 = {}, acc1 = {};
    const float* erow = &Es[jt * 16 + l16][0];
#pragma unroll 4
    for (int kk = 0; kk < DE / 4; ++kk) {
      const int k0 = kk * 4 + lhalf * 2;
      v2f a, bf0, bf1;
      a.x = erow[k0];
      a.y = erow[k0 + 1];
      bf0.x = Wp0s[l16][k0];
      bf0.y = Wp0s[l16][k0 + 1];
      bf1.x = Wp0s[16 + l16][k0];
      bf1.y = Wp0s[16 + l16][k0 + 1];
      acc0 = wmma_f32_k4(a, bf0, acc0);
      acc1 = wmma_f32_k4(a, bf1, acc1);
    }
    const float bias0 = bp0[l16], bias1 = bp0[16 + l16];
#pragma unroll
    for (int r = 0; r < 8; ++r) {
      const int j = jt * 16 + r + lhalf * 8;
      const float bm = bmrow[j];
      p0s[j][l16] = tanhf(acc0[r] + bias0) * bm;
      p0s[j][16 + l16] = tanhf(acc1[r] + bias1) * bm;
    }
  }
  __syncthreads();

  // scores[j,h] = (1/sqrt(DF)) * sum_d p0[j,d] * Q[i,h,d] * K[j,h,d]
  const float inv_sqrt_df = 0.17677669529663687f;
  for (int p = tid; p < NN * NH; p += 256) {
    const int j = p >> 3, h = p & 7;
    const float* krow = K + ((size_t)b * NN + j) * DX + h * DF;
    float s = 0.f;
#pragma unroll 8
    for (int d = 0; d < DF; ++d) s += p0s[j][d] * Qs[h][d] * krow[d];
    scs[j * 9 + h] = s * inv_sqrt_df;
  }
  __syncthreads();

  // softmax over j for head h == wave; only the diagonal weight is needed
  {
    float mx = -3.4e38f;
    for (int j = lane; j < NN; j += 32) mx = fmaxf(mx, scs[j * 9 + wave]);
#pragma unroll
    for (int off = 16; off; off >>= 1) mx = fmaxf(mx, __shfl_xor(mx, off, 32));
    float sm = 0.f;
    for (int j = lane; j < NN; j += 32) sm += expf(scs[j * 9 + wave] - mx);
#pragma unroll
    for (int off = 16; off; off >>= 1) sm += __shfl_xor(sm, off, 32);
    if (lane == 0) adiag[wave] = expf(scs[i * 9 + wave] - mx) / sm;
  }

  // p1 on the diagonal only: p1d[d] = tanh(E[b,i,i,:] @ Wp1[d,:] + bp1[d]) * bm
  if (tid < DF) {
    const float* er = &Es[i][0];
    const float* wr = Wp1 + tid * DE;
    float s = bp1[tid];
#pragma unroll 8
    for (int k = 0; k < DE; ++k) s += er[k] * wr[k];
    p1d[tid] = tanhf(s) * bmrow[i];
  }
  __syncthreads();

  // weighted + residual -> LN1 -> Wo + ReLU + residual -> LN2
  const size_t rowoff = ((size_t)b * NN + i) * DX;
  const float xv = X[rowoff + tid];
  const float val = adiag[tid >> 5] * p1d[tid & 31] * V[rowoff + tid] + xv;
  const float mu = blockSum256(val, red) * (1.f / DX);
  const float dfv = val - mu;
  const float var = blockSum256(dfv * dfv, red) * (1.f / DX);
  y1[tid] = dfv * rsqrtf(var + 1e-5f) * g1[tid] + b1[tid];
  __syncthreads();

  float s2 = bo[tid];
  const float* wrow = Wo + (size_t)tid * DX;
#pragma unroll 8
  for (int k = 0; k < DX; ++k) s2 += y1[k] * wrow[k];
  const float o = fmaxf(s2, 0.f) + xv;
  const float mu2 = blockSum256(o, red) * (1.f / DX);
  const float d2 = o - mu2;
  const float var2 = blockSum256(d2 * d2, red) * (1.f / DX);
  out[rowoff + tid] = d2 * rsqrtf(var2 + 1e-5f) * g2[tid] + b2[tid];
}

// ---------------------------------------------------------------------------
// Kernel 3: SE layer, one block per batch, in-place scale of d_out.
// ---------------------------------------------------------------------------
__global__ __launch_bounds__(256) void se_kernel(
    const float* __restrict__ Wse1, const float* __restrict__ bse1,
    const float* __restrict__ Wse2, const float* __restrict__ bse2,
    float* __restrict__ out) {
  const int b = blockIdx.x;
  const int c = threadIdx.x;
  __shared__ float mean_s[DX];
  __shared__ float hid[DX / 16];
  float* ob = out + (size_t)b * NN * DX;

  float s = 0.f;
  for (int i = 0; i < NN; ++i) s += ob[(size_t)i * DX + c];
  mean_s[c] = s * (1.f / NN);
  __syncthreads();

  if (c < DX / 16) {
    const float* wr = Wse1 + c * DX;
    float t = bse1[c];
    for (int k = 0; k < DX; ++k) t += mean_s[k] * wr[k];
    hid[c] = fmaxf(t, 0.f);
  }
  __syncthreads();

  float t = bse2[c];
  const float* wr = Wse2 + c * (DX / 16);
#pragma unroll
  for (int k = 0; k < DX / 16; ++k) t += hid[k] * wr[k];
  const float se = 1.f / (1.f + expf(-t));
  for (int i = 0; i < NN; ++i) ob[(size_t)i * DX + c] *= se;
}

extern "C" void kernel_launch(void* const* d_in, const int* in_sizes, int n_in,
                              void* d_out, int out_size, void* d_ws,
                              size_t ws_size, hipStream_t stream) {
  (void)in_sizes; (void)n_in; (void)out_size; (void)ws_size;
  const float* X = (const float*)d_in[0];
  const float* E = (const float*)d_in[1];
  const float* amask = (const float*)d_in[2];
  const float* bmask = (const float*)d_in[3];
  const float* Wq = (const float*)d_in[4];  const float* bq = (const float*)d_in[5];
  const float* Wk = (const float*)d_in[6];  const float* bk = (const float*)d_in[7];
  const float* Wv = (const float*)d_in[8];  const float* bv = (const float*)d_in[9];
  const float* Wp0 = (const float*)d_in[10]; const float* bp0 = (const float*)d_in[11];
  const float* Wp1 = (const float*)d_in[12]; const float* bp1 = (const float*)d_in[13];
  const float* g1 = (const float*)d_in[14]; const float* b1 = (const float*)d_in[15];
  const float* Wo = (const float*)d_in[16]; const float* bo = (const float*)d_in[17];
  const float* g2 = (const float*)d_in[18]; const float* b2 = (const float*)d_in[19];
  const float* Wse1 = (const float*)d_in[20]; const float* bse1 = (const float*)d_in[21];
  const float* Wse2 = (const float*)d_in[22]; const float* bse2 = (const float*)d_in[23];

  float* Qw = (float*)d_ws;
  float* Kw = Qw + (size_t)BS * NN * DX;
  float* Vw = Kw + (size_t)BS * NN * DX;
  float* out = (float*)d_out;

  qkv_kernel<<<dim3(BS * (NN / 16)), 256, 0, stream>>>(
      X, amask, Wq, bq, Wk, bk, Wv, bv, Qw, Kw, Vw);
  attn_kernel<<<dim3(NN, BS), 256, 0, stream>>>(
      X, E, bmask, Qw, Kw, Vw, Wp0, bp0, Wp1, bp1, g1, b1, Wo, bo, g2, b2, out);
  se_kernel<<<dim3(BS), 256, 0, stream>>>(Wse1, bse1, Wse2, bse2, out);
}